// ModelStructure_72069551227181
// MI455X (gfx1250) — compile-verified
//
#include <hip/hip_runtime.h>
#include <hip/hip_bf16.h>
#include <stdint.h>

typedef __attribute__((ext_vector_type(2))) float v2f;
typedef __attribute__((ext_vector_type(8))) float v8f;

#define TM 128
#define TN 64
#define TK 16
#define APAD 4   // A row pad -> stride 20 floats (80B, 16B aligned)
#define BPAD 4   // B row pad -> stride 68 floats (272B, 16B aligned)

// Issue one 16-byte async global->LDS copy (CDNA5 TDM-lite path, ASYNCcnt).
__device__ __forceinline__ void async_copy16(uint32_t lds_off, const void* gsrc) {
    asm volatile("global_load_async_to_lds_b128 %0, %1, off"
                 :: "v"(lds_off), "v"((uint64_t)(uintptr_t)gsrc)
                 : "memory");
}
__device__ __forceinline__ void wait_async0() {
    asm volatile("s_wait_asynccnt 0x0" ::: "memory");
}

// ---------------------------------------------------------------------------
// GEMM: C[M,N] = A[M,K] @ W[K,N] + bias, fp32 via V_WMMA_F32_16X16X4_F32.
// 256 threads = 8 waves; each wave computes a 32x32 tile (2x2 WMMA frags);
// block tile 128(M) x 64(N); K staged in 16-chunks, double-buffered in LDS
// via global_load_async_to_lds_b128 + s_wait_asynccnt.
// ---------------------------------------------------------------------------
__global__ __launch_bounds__(256)
void gemm_bias_f32(const float* __restrict__ A, const float* __restrict__ W,
                   const float* __restrict__ bias, float* __restrict__ C,
                   int M, int N, int K) {
    __shared__ float As[2][TM][TK + APAD];
    __shared__ float Bs[2][TK][TN + BPAD];

    const int tid  = threadIdx.x;
    const int lane = tid & 31;
    const int wave = tid >> 5;
    const int wm   = (wave & 3) * 32;   // wave M offset inside block tile
    const int wn   = (wave >> 2) * 32;  // wave N offset inside block tile
    const int m0   = blockIdx.y * TM;
    const int n0   = blockIdx.x * TN;
    const int r    = lane & 15;
    const int koff = (lane >> 4) << 1;  // 0 or 2 (ISA f32 A/B fragment layout)
    const int rsel = (lane >> 4) << 3;  // 0 or 8 (ISA f32 C fragment layout)

    // per-thread staging coordinates
    const int arow0 = tid >> 2;             // 0..63   (A chunk 0)
    const int aq0   = (tid & 3) << 2;       // 0,4,8,12
    const int arow1 = (tid + 256) >> 2;     // 64..127 (A chunk 1)
    const int brow  = tid >> 4;             // 0..15
    const int bq    = (tid & 15) << 2;      // 0..60

    auto stage = [&](int kt, int bsel) {
        // A tile (128x16): two b128 async copies per thread
        async_copy16((uint32_t)(uintptr_t)&As[bsel][arow0][aq0],
                     A + (size_t)(m0 + arow0) * K + kt + aq0);
        async_copy16((uint32_t)(uintptr_t)&As[bsel][arow1][aq0],
                     A + (size_t)(m0 + arow1) * K + kt + aq0);
        // B tile (16x64): one b128 async copy per thread, guarded tail
        int col = n0 + bq;
        const float* src = W + (size_t)(kt + brow) * N + col;
        if (col + 3 < N) {
            async_copy16((uint32_t)(uintptr_t)&Bs[bsel][brow][bq], src);
        } else {
#pragma unroll
            for (int e = 0; e < 4; ++e)
                Bs[bsel][brow][bq + e] = (col + e < N) ? src[e] : 0.0f;
        }
    };

    v8f acc[2][2];
#pragma unroll
    for (int i = 0; i < 2; ++i)
#pragma unroll
        for (int j = 0; j < 2; ++j)
            acc[i][j] = v8f{0.f, 0.f, 0.f, 0.f, 0.f, 0.f, 0.f, 0.f};

    stage(0, 0);
    int buf = 0;
    for (int kt = 0; kt < K; kt += TK) {
        wait_async0();          // this wave's async copies into `buf` done
        __syncthreads();        // all waves' copies visible
        if (kt + TK < K) stage(kt + TK, buf ^ 1);   // prefetch next tile

#pragma unroll
        for (int kk = 0; kk < TK; kk += 4) {
            v2f af[2], bf[2];
#pragma unroll
            for (int i = 0; i < 2; ++i) {
                int row = wm + i * 16 + r;
                af[i].x = As[buf][row][kk + koff];
                af[i].y = As[buf][row][kk + koff + 1];
            }
#pragma unroll
            for (int j = 0; j < 2; ++j) {
                int col = wn + j * 16 + r;
                bf[j].x = Bs[buf][kk + koff][col];
                bf[j].y = Bs[buf][kk + koff + 1][col];
            }
#pragma unroll
            for (int i = 0; i < 2; ++i)
#pragma unroll
                for (int j = 0; j < 2; ++j)
                    acc[i][j] = __builtin_amdgcn_wmma_f32_16x16x4_f32(
                        false, af[i], false, bf[j], (short)0, acc[i][j],
                        false, false);
        }
        __syncthreads();        // all reads of `buf` done before it's restaged
        buf ^= 1;
    }

    // ---- epilogue: bias add + store (guarded on N)
#pragma unroll
    for (int i = 0; i < 2; ++i)
#pragma unroll
        for (int j = 0; j < 2; ++j) {
            int colg = n0 + wn + j * 16 + r;
            if (colg < N) {
                float bv = bias[colg];
#pragma unroll
                for (int v = 0; v < 8; ++v) {
                    int rowg = m0 + wm + i * 16 + v + rsel;
                    C[(size_t)rowg * N + colg] = acc[i][j][v] + bv;
                }
            }
        }
}

// ---------------------------------------------------------------------------
// Column sums / sums-of-squares for BatchNorm (training-mode batch stats).
// ---------------------------------------------------------------------------
__global__ __launch_bounds__(256)
void col_stats(const float* __restrict__ Z, float* __restrict__ cs,
               float* __restrict__ css, int M, int N) {
    int c = blockIdx.x * 256 + threadIdx.x;
    if (c >= N) return;
    int row0 = blockIdx.y * 256;
    float s = 0.f, ss = 0.f;
    for (int rr = 0; rr < 256; ++rr) {
        float v = Z[(size_t)(row0 + rr) * N + c];
        s += v; ss += v * v;
    }
    atomicAdd(&cs[c], s);
    atomicAdd(&css[c], ss);
}

// ---------------------------------------------------------------------------
// BN normalize (biased var, eps=1e-5) + activation in place.
// leaky==0 -> ReLU, leaky==1 -> leaky ReLU slope 0.01.
// ---------------------------------------------------------------------------
__global__ __launch_bounds__(256)
void bn_act(float* __restrict__ Z, const float* __restrict__ cs,
            const float* __restrict__ css, const float* __restrict__ gam,
            const float* __restrict__ bet, int M, int N, int leaky) {
    size_t total = (size_t)M * N;
    float invM = 1.0f / (float)M;
    for (size_t idx = (size_t)blockIdx.x * blockDim.x + threadIdx.x;
         idx < total; idx += (size_t)gridDim.x * blockDim.x) {
        int c = (int)(idx % (size_t)N);
        float mean = cs[c] * invM;
        float var  = css[c] * invM - mean * mean;
        float z = Z[idx];
        float y = (z - mean) * rsqrtf(var + 1e-5f) * gam[c] + bet[c];
        Z[idx] = y > 0.0f ? y : (leaky ? 0.01f * y : 0.0f);
    }
}

// ---------------------------------------------------------------------------
// Per (pair, node) Hotelling T^2:
//   Sp = (Xic^T Xic + Xjc^T Xjc) / 254  (64x64, K=128, WMMA f32)
//   solve Sp y = (mi - mj) via in-LDS Cholesky, t2 = 64 * d.y, ReLU.
// grid = (128 nodes, 6 pairs), block = 256 threads (8 waves, 2 tiles each).
// ---------------------------------------------------------------------------
#define XS 68   // sX row stride (floats): 272B, 16B-aligned for async b128
__global__ __launch_bounds__(256)
void cov_t2(const float* __restrict__ gdt, float* __restrict__ t2out) {
    __shared__ float sX[128 * XS];   // one centered 128x64 slice (padded)
    __shared__ float sS[64 * 65];    // pooled covariance (padded)
    __shared__ float sv[5 * 64];     // mi | mj | d | z | y

    const int n = blockIdx.x;
    const int p = blockIdx.y;
    const int PI[6] = {0, 0, 0, 1, 1, 2};
    const int PJ[6] = {1, 2, 3, 2, 3, 3};
    const int gsel[2] = {PI[p], PJ[p]};

    const int tid  = threadIdx.x;
    const int lane = tid & 31;
    const int wave = tid >> 5;
    const int r    = lane & 15;
    const int koff = (lane >> 4) << 1;
    const int rsel = (lane >> 4) << 3;

    // two 16x16 output tiles per wave out of the 4x4 tile grid
    int m0s[2], n0s[2];
    v8f acc[2];
#pragma unroll
    for (int t = 0; t < 2; ++t) {
        int tl = wave + t * 8;
        m0s[t] = (tl >> 2) * 16;
        n0s[t] = (tl & 3) * 16;
        acc[t] = v8f{0.f, 0.f, 0.f, 0.f, 0.f, 0.f, 0.f, 0.f};
    }

    for (int half = 0; half < 2; ++half) {
        const float* Xh = gdt + (size_t)gsel[half] * 128 * 128 * 64 +
                          (size_t)n * 64;
        // async-load 128x64 slice (row stride N*D = 8192 floats), b128 chunks
        for (int t = tid; t < 128 * 16; t += 256) {
            int b = t >> 4, dq = (t & 15) << 2;
            async_copy16((uint32_t)(uintptr_t)&sX[b * XS + dq],
                         Xh + (size_t)b * 8192 + dq);
        }
        wait_async0();
        __syncthreads();
        // column means
        if (tid < 64) {
            float s = 0.f;
            for (int b = 0; b < 128; ++b) s += sX[b * XS + tid];
            sv[half * 64 + tid] = s * (1.0f / 128.0f);
        }
        __syncthreads();
        // center
        for (int t = tid; t < 128 * 64; t += 256) {
            int b = t >> 6, d = t & 63;
            sX[b * XS + d] -= sv[half * 64 + d];
        }
        __syncthreads();
        // accumulate Xc^T Xc into acc via WMMA f32 (M=d, N=d, K=b)
#pragma unroll
        for (int t = 0; t < 2; ++t) {
            for (int k = 0; k < 128; k += 4) {
                v2f a, b;
                a.x = sX[(k + koff) * XS + m0s[t] + r];
                a.y = sX[(k + koff + 1) * XS + m0s[t] + r];
                b.x = sX[(k + koff) * XS + n0s[t] + r];
                b.y = sX[(k + koff + 1) * XS + n0s[t] + r];
                acc[t] = __builtin_amdgcn_wmma_f32_16x16x4_f32(
                    false, a, false, b, (short)0, acc[t], false, false);
            }
        }
        __syncthreads();  // before sX is overwritten / reused
    }

    // d = mi - mj
    if (tid < 64) sv[128 + tid] = sv[tid] - sv[64 + tid];
    // write pooled covariance Sp = sum / (2*(B-1)) = sum / 254
#pragma unroll
    for (int t = 0; t < 2; ++t)
#pragma unroll
        for (int v = 0; v < 8; ++v)
            sS[(m0s[t] + v + rsel) * 65 + n0s[t] + r] =
                acc[t][v] * (1.0f / 254.0f);
    __syncthreads();

    // ---- Cholesky Sp = L L^T (lower, in place) ----
    for (int k = 0; k < 64; ++k) {
        if (tid == 0) sS[k * 65 + k] = sqrtf(sS[k * 65 + k]);
        __syncthreads();
        if (tid > k && tid < 64) sS[tid * 65 + k] /= sS[k * 65 + k];
        __syncthreads();
        if (tid > k && tid < 64) {
            float ljk = sS[tid * 65 + k];
            for (int i = tid; i < 64; ++i)
                sS[i * 65 + tid] -= sS[i * 65 + k] * ljk;
        }
        __syncthreads();
    }
    // ---- forward solve L z = d ----
    if (tid < 64) sv[192 + tid] = sv[128 + tid];
    __syncthreads();
    for (int k = 0; k < 64; ++k) {
        if (tid == 0) sv[192 + k] /= sS[k * 65 + k];
        __syncthreads();
        if (tid > k && tid < 64) sv[192 + tid] -= sS[tid * 65 + k] * sv[192 + k];
        __syncthreads();
    }
    // ---- back solve L^T y = z ----
    if (tid < 64) sv[256 + tid] = sv[192 + tid];
    __syncthreads();
    for (int k = 63; k >= 0; --k) {
        if (tid == 0) sv[256 + k] /= sS[k * 65 + k];
        __syncthreads();
        if (tid < k) sv[256 + tid] -= sS[k * 65 + tid] * sv[256 + k];
        __syncthreads();
    }
    if (tid == 0) {
        float dot = 0.f;
        for (int d = 0; d < 64; ++d) dot += sv[128 + d] * sv[256 + d];
        float t2v = 64.0f * dot;   // B*B/(2B) = 64
        t2out[p * 128 + n] = t2v > 0.f ? t2v : 0.f;
    }
}

// ---------------------------------------------------------------------------
// Final reductions over t2 [6,128].
// ---------------------------------------------------------------------------
__global__ __launch_bounds__(128)
void finalize(const float* __restrict__ t2, const float* __restrict__ Wm,
              float* __restrict__ tg, float* __restrict__ tnode,
              float* __restrict__ tmin, float* __restrict__ intra) {
    __shared__ float sT[6 * 128];
    __shared__ float sR[128];
    int tid = threadIdx.x;
    for (int i = tid; i < 6 * 128; i += 128) sT[i] = t2[i];
    __syncthreads();
    {
        float s = 0.f, mn = 3.4e38f;
        for (int p = 0; p < 6; ++p) {
            float v = sT[p * 128 + tid];
            s += v; mn = fminf(mn, v);
        }
        float mean = s * (1.0f / 6.0f);
        float ia = 0.f;
        for (int p = 0; p < 6; ++p) {
            float v = sT[p * 128 + tid];
            ia += fabsf(v - mean) * mean;
        }
        tnode[tid] = s; tmin[tid] = mn; intra[tid] = ia;
    }
    for (int p = 0; p < 6; ++p) {
        float a = 0.f;
        for (int nn = 0; nn < 128; ++nn)
            a += sT[p * 128 + nn] * Wm[nn * 128 + tid];
        sR[tid] = a > 0.f ? a : 0.f;
        __syncthreads();
        for (int off = 64; off > 0; off >>= 1) {
            if (tid < off) sR[tid] += sR[tid + off];
            __syncthreads();
        }
        if (tid == 0) tg[p] = sR[0];
        __syncthreads();
    }
}

// ---------------------------------------------------------------------------
extern "C" void kernel_launch(void* const* d_in, const int* in_sizes, int n_in,
                              void* d_out, int out_size, void* d_ws, size_t ws_size,
                              hipStream_t stream) {
    (void)in_sizes; (void)n_in; (void)out_size; (void)ws_size;
    const float* xg  = (const float*)d_in[0];
    const float* W1  = (const float*)d_in[1];
    const float* b1  = (const float*)d_in[2];
    const float* g1  = (const float*)d_in[3];
    const float* be1 = (const float*)d_in[4];
    const float* W2  = (const float*)d_in[5];
    const float* b2  = (const float*)d_in[6];
    const float* g2  = (const float*)d_in[7];
    const float* be2 = (const float*)d_in[8];
    const float* W3  = (const float*)d_in[9];
    const float* b3  = (const float*)d_in[10];
    const float* g3  = (const float*)d_in[11];
    const float* be3 = (const float*)d_in[12];
    const float* W4  = (const float*)d_in[13];
    const float* b4  = (const float*)d_in[14];
    const float* g4  = (const float*)d_in[15];
    const float* be4 = (const float*)d_in[16];
    const float* Wm  = (const float*)d_in[17];
    // d_in[18] = transform_flag (fixed 1 in setup)

    float* out  = (float*)d_out;
    float* ws   = (float*)d_ws;
    float* bufA = ws;                                   // 16384*1024
    float* bufB = ws + (size_t)16384 * 1024;            // 16384*2000
    float* cs   = bufB + (size_t)16384 * 2000;          // 2048
    float* css  = cs + 2048;                            // 2048
    float* t2   = css + 2048;                           // 6*128

    const size_t GDT = (size_t)4 * 128 * 128 * 64;      // 4194304
    float* out_tg    = out + GDT;
    float* out_node  = out_tg + 6;
    float* out_min   = out_node + 128;
    float* out_intra = out_min + 128;

    const int M = 16384;
    auto stats_bn = [&](float* Z, int N, const float* gam, const float* bet,
                        int leaky) {
        hipMemsetAsync(cs, 0, 2 * 2048 * sizeof(float), stream);
        dim3 gs((N + 255) / 256, M / 256);
        col_stats<<<gs, 256, 0, stream>>>(Z, cs, css, M, N);
        bn_act<<<2048, 256, 0, stream>>>(Z, cs, css, gam, bet, M, N, leaky);
    };

    for (int grp = 0; grp < 4; ++grp) {
        const float* Xg = xg + (size_t)grp * 128 * 128 * 512;
        float* gout     = out + (size_t)grp * 128 * 128 * 64;

        gemm_bias_f32<<<dim3(16, 128), 256, 0, stream>>>(Xg, W1, b1, bufA,
                                                         M, 1024, 512);
        stats_bn(bufA, 1024, g1, be1, 0);
        gemm_bias_f32<<<dim3(32, 128), 256, 0, stream>>>(bufA, W2, b2, bufB,
                                                         M, 2000, 1024);
        stats_bn(bufB, 2000, g2, be2, 0);
        gemm_bias_f32<<<dim3(16, 128), 256, 0, stream>>>(bufB, W3, b3, bufA,
                                                         M, 1024, 2000);
        stats_bn(bufA, 1024, g3, be3, 0);
        gemm_bias_f32<<<dim3(1, 128), 256, 0, stream>>>(bufA, W4, b4, gout,
                                                        M, 64, 1024);
        stats_bn(gout, 64, g4, be4, 1);
    }

    cov_t2<<<dim3(128, 6), 256, 0, stream>>>(out, t2);
    finalize<<<1, 128, 0, stream>>>(t2, Wm, out_tg, out_node, out_min,
                                    out_intra);
}